// Attention_Block_17686675325259
// MI455X (gfx1250) — compile-verified
//
#include <hip/hip_runtime.h>
#include <hip/hip_bf16.h>
#include <math.h>

// Problem constants (match reference)
#define NN   50000
#define EE   800000
#define DDIM 128
#define HH   8
#define DHH  16
#define TILES (NN / 16)          // 3125 exact
#define LN_EPS 1e-5f

typedef __attribute__((ext_vector_type(2))) float v2f;
typedef __attribute__((ext_vector_type(8))) float v8f;

// ---------------------------------------------------------------------------
// WMMA helper: one wave computes a 16x16 f32 tile of  As(16x128) @ W(128x128)
// restricted to columns [colbase, colbase+16).  Uses V_WMMA_F32_16X16X4_F32.
// A-frag layout (ISA 7.12.2, 32-bit A 16x4): lane L holds row M=L&15,
//   VGPR0 = K=koff, VGPR1 = K=koff+1 where koff = 2*(L>>4).
// B-frag layout mirrors: lane L holds col N=L&15, rows K=koff / koff+1.
// C/D layout: reg r -> row r + 8*(L>>4), col = L&15 (+colbase).
// ---------------------------------------------------------------------------
__device__ __forceinline__ v8f wmma_gemm_frag(const float* As,
                                              const float* __restrict__ W,
                                              int col, int lane) {
  const int row  = lane & 15;
  const int koff = (lane >> 4) << 1;
  v8f acc = {};
#pragma unroll
  for (int k0 = 0; k0 < DDIM; k0 += 4) {
    v2f a, b;
    a.x = As[row * DDIM + k0 + koff];
    a.y = As[row * DDIM + k0 + koff + 1];
    b.x = W[(k0 + koff) * DDIM + col];
    b.y = W[(k0 + koff + 1) * DDIM + col];
    acc = __builtin_amdgcn_wmma_f32_16x16x4_f32(
        /*neg_a=*/false, a, /*neg_b=*/false, b,
        /*c_mod=*/(short)0, acc, /*reuse_a=*/false, /*reuse_b=*/false);
  }
  return acc;
}

// ---------------------------------------------------------------------------
// Async tile stage: 16x128 f32 tile (8KB) global -> LDS via CDNA5
// GLOBAL_LOAD_ASYNC_TO_LDS_B128 (ASYNCcnt-tracked, no VGPR round-trip).
// LDS byte address for VDST = low 32 bits of the generic pointer
// (LDS aperture truncation, ISA sec 10.2).  Caller must issue
// wait_async_lds() before __syncthreads().
// ---------------------------------------------------------------------------
__device__ __forceinline__ void load_tile16x128_async(const float* __restrict__ src,
                                                      int tile, float* dst, int tid) {
  const float4* s4 = (const float4*)(src + (size_t)tile * 16 * DDIM);
#pragma unroll
  for (int i = tid; i < 512; i += 256) {
    unsigned lds_addr = (unsigned)(size_t)(const void*)(dst + 4 * i);
    unsigned long long gaddr = (unsigned long long)(size_t)(const void*)(s4 + i);
    asm volatile("global_load_async_to_lds_b128 %0, %1, off"
                 :: "v"(lds_addr), "v"(gaddr)
                 : "memory");
  }
}

__device__ __forceinline__ void wait_async_lds() {
  asm volatile("s_wait_asynccnt 0" ::: "memory");
}

// sign-aware float atomic max (init value must be -FLT_MAX)
__device__ __forceinline__ void atomicMaxFloat(float* addr, float val) {
  if (__float_as_uint(val) >> 31)
    atomicMin((unsigned int*)addr, __float_as_uint(val));
  else
    atomicMax((int*)addr, __float_as_int(val));
}

// ---------------------------------------------------------------------------
// Kernel 1: init segment-max / denom / agg accumulators
// ---------------------------------------------------------------------------
__global__ void k_init(float* __restrict__ m, float* __restrict__ denom,
                       float* __restrict__ agg) {
  int i = blockIdx.x * 256 + threadIdx.x;
  if (i < NN * HH) { m[i] = -3.402823466e38f; denom[i] = 0.0f; }
  if (i < NN * DDIM) agg[i] = 0.0f;
}

// ---------------------------------------------------------------------------
// Kernel 2: Q/K/V projection.  1 block = 16 rows; 8 waves, wave w owns cols
// [16w,16w+16) of all three outputs.
// ---------------------------------------------------------------------------
__global__ void __launch_bounds__(256)
k_qkv(const float* __restrict__ x,
      const float* __restrict__ Wq, const float* __restrict__ bq,
      const float* __restrict__ Wk, const float* __restrict__ bk,
      const float* __restrict__ Wv, const float* __restrict__ bv,
      float* __restrict__ q, float* __restrict__ k, float* __restrict__ v) {
  __shared__ float sx[16 * DDIM];
  const int tile = blockIdx.x;
  const int tid  = threadIdx.x;
  // warm L2/WGP$ for the (tiny, fully reused) weight matrices
  __builtin_prefetch(Wq, 0, 3);
  __builtin_prefetch(Wk, 0, 3);
  __builtin_prefetch(Wv, 0, 3);
  load_tile16x128_async(x, tile, sx, tid);
  wait_async_lds();
  __syncthreads();

  const int w = tid >> 5, lane = tid & 31;
  const int col = w * 16 + (lane & 15);
  const int hi  = (lane >> 4) << 3;

  v8f aq = wmma_gemm_frag(sx, Wq, col, lane);
  {
    float bb = bq[col];
#pragma unroll
    for (int r = 0; r < 8; ++r) {
      int grow = tile * 16 + r + hi;
      q[(size_t)grow * DDIM + col] = aq[r] + bb;
    }
  }
  v8f ak = wmma_gemm_frag(sx, Wk, col, lane);
  {
    float bb = bk[col];
#pragma unroll
    for (int r = 0; r < 8; ++r) {
      int grow = tile * 16 + r + hi;
      k[(size_t)grow * DDIM + col] = ak[r] + bb;
    }
  }
  v8f av = wmma_gemm_frag(sx, Wv, col, lane);
  {
    float bb = bv[col];
#pragma unroll
    for (int r = 0; r < 8; ++r) {
      int grow = tile * 16 + r + hi;
      v[(size_t)grow * DDIM + col] = av[r] + bb;
    }
  }
}

// ---------------------------------------------------------------------------
// Kernel 3: per-(edge,head) attention logits + segment max
// ---------------------------------------------------------------------------
__global__ void k_scores(const float* __restrict__ q, const float* __restrict__ k,
                         const int* __restrict__ idx_i, const int* __restrict__ idx_j,
                         float* __restrict__ e, float* __restrict__ m) {
  int t = blockIdx.x * 256 + threadIdx.x;
  if (t >= EE * HH) return;
  int eid = t >> 3, h = t & 7;
  int i = idx_i[eid], j = idx_j[eid];
  const float4* qi = (const float4*)(q + (size_t)i * DDIM + h * DHH);
  const float4* kj = (const float4*)(k + (size_t)j * DDIM + h * DHH);
  float s = 0.0f;
#pragma unroll
  for (int c = 0; c < 4; ++c) {
    float4 a = qi[c], b = kj[c];
    s += a.x * b.x + a.y * b.y + a.z * b.z + a.w * b.w;
  }
  s *= 0.25f;  // 1/sqrt(DH), DH=16
  e[t] = s;
  atomicMaxFloat(&m[i * HH + h], s);
}

// ---------------------------------------------------------------------------
// Kernel 4: a = exp(e-m); scatter-add denom and unnormalized agg
// ---------------------------------------------------------------------------
__global__ void k_accum(const float* __restrict__ e, const float* __restrict__ m,
                        const float* __restrict__ v,
                        const int* __restrict__ idx_i, const int* __restrict__ idx_j,
                        float* __restrict__ denom, float* __restrict__ agg) {
  int t = blockIdx.x * 256 + threadIdx.x;
  if (t >= EE * HH) return;
  int eid = t >> 3, h = t & 7;
  int i = idx_i[eid], j = idx_j[eid];
  float a = __expf(e[t] - m[i * HH + h]);
  atomicAdd(&denom[i * HH + h], a);
  const float* vj = v + (size_t)j * DDIM + h * DHH;
  float* ag = agg + (size_t)i * DDIM + h * DHH;
#pragma unroll
  for (int d = 0; d < DHH; ++d) atomicAdd(&ag[d], a * vj[d]);
}

// row LayerNorm over 128 cols, one wave handles one row (4 elems/lane)
__device__ __forceinline__ void row_layernorm_store(
    const float* srow, const float* __restrict__ g, const float* __restrict__ b,
    float* __restrict__ dst, int lane) {
  float4 vv = ((const float4*)srow)[lane];
  float s  = vv.x + vv.y + vv.z + vv.w;
  float sq = vv.x * vv.x + vv.y * vv.y + vv.z * vv.z + vv.w * vv.w;
#pragma unroll
  for (int off = 16; off > 0; off >>= 1) {
    s  += __shfl_xor(s, off, 32);
    sq += __shfl_xor(sq, off, 32);
  }
  float mean = s * (1.0f / 128.0f);
  float var  = sq * (1.0f / 128.0f) - mean * mean;
  float rstd = rsqrtf(var + LN_EPS);
  int c = lane * 4;
  dst[c + 0] = (vv.x - mean) * rstd * g[c + 0] + b[c + 0];
  dst[c + 1] = (vv.y - mean) * rstd * g[c + 1] + b[c + 1];
  dst[c + 2] = (vv.z - mean) * rstd * g[c + 2] + b[c + 2];
  dst[c + 3] = (vv.w - mean) * rstd * g[c + 3] + b[c + 3];
}

// ---------------------------------------------------------------------------
// Kernel 5: normalize agg, output projection (WMMA), +x residual, LayerNorm1
// ---------------------------------------------------------------------------
__global__ void __launch_bounds__(256)
k_attn_out(const float* __restrict__ x, const float* __restrict__ agg,
           const float* __restrict__ denom, const float* __restrict__ Wo,
           const float* __restrict__ bo, const float* __restrict__ g1,
           const float* __restrict__ be1, float* __restrict__ h_out) {
  __shared__ float sA[16 * DDIM];
  const int tile = blockIdx.x;
  const int tid  = threadIdx.x;
  __builtin_prefetch(Wo, 0, 3);
  // normalized agg tile -> LDS (guard zero in-degree nodes)
  for (int idx = tid; idx < 16 * DDIM; idx += 256) {
    int r = idx >> 7, c = idx & 127;
    int grow = tile * 16 + r;
    float dn = denom[grow * HH + (c >> 4)];
    float val = agg[(size_t)grow * DDIM + c];
    sA[idx] = (dn > 0.0f) ? val / dn : 0.0f;
  }
  __syncthreads();

  const int w = tid >> 5, lane = tid & 31;
  const int col = w * 16 + (lane & 15);
  const int hi  = (lane >> 4) << 3;
  v8f acc = wmma_gemm_frag(sA, Wo, col, lane);
  __syncthreads();  // everyone done reading sA before overwrite
  float bb = bo[col];
#pragma unroll
  for (int r = 0; r < 8; ++r) {
    int row = r + hi;
    int grow = tile * 16 + row;
    sA[row * DDIM + col] = acc[r] + bb + x[(size_t)grow * DDIM + col];
  }
  __syncthreads();
  // 8 waves x 2 rows each
#pragma unroll
  for (int rr = 0; rr < 2; ++rr) {
    int row = w * 2 + rr;
    int grow = tile * 16 + row;
    row_layernorm_store(sA + row * DDIM, g1, be1,
                        h_out + (size_t)grow * DDIM, lane);
  }
}

// ---------------------------------------------------------------------------
// Kernel 6: FFN (two WMMA GEMMs with ReLU) + residual + LayerNorm2 -> d_out
// ---------------------------------------------------------------------------
__global__ void __launch_bounds__(256)
k_ffn(const float* __restrict__ h_in,
      const float* __restrict__ W1, const float* __restrict__ bb1,
      const float* __restrict__ W2, const float* __restrict__ bb2,
      const float* __restrict__ g2, const float* __restrict__ be2,
      float* __restrict__ out) {
  __shared__ float sH[16 * DDIM];
  __shared__ float sT[16 * DDIM];
  const int tile = blockIdx.x;
  const int tid  = threadIdx.x;
  __builtin_prefetch(W1, 0, 3);
  __builtin_prefetch(W2, 0, 3);
  load_tile16x128_async(h_in, tile, sH, tid);
  wait_async_lds();
  __syncthreads();

  const int w = tid >> 5, lane = tid & 31;
  const int col = w * 16 + (lane & 15);
  const int hi  = (lane >> 4) << 3;

  v8f t1 = wmma_gemm_frag(sH, W1, col, lane);
  float b1 = bb1[col];
#pragma unroll
  for (int r = 0; r < 8; ++r) {
    int row = r + hi;
    sT[row * DDIM + col] = fmaxf(t1[r] + b1, 0.0f);  // ReLU
  }
  __syncthreads();

  v8f t2 = wmma_gemm_frag(sT, W2, col, lane);
  __syncthreads();  // all reads of sT done before overwrite
  float b2 = bb2[col];
#pragma unroll
  for (int r = 0; r < 8; ++r) {
    int row = r + hi;
    sT[row * DDIM + col] = t2[r] + b2 + sH[row * DDIM + col];  // residual
  }
  __syncthreads();
#pragma unroll
  for (int rr = 0; rr < 2; ++rr) {
    int row = w * 2 + rr;
    int grow = tile * 16 + row;
    row_layernorm_store(sT + row * DDIM, g2, be2,
                        out + (size_t)grow * DDIM, lane);
  }
}

// ---------------------------------------------------------------------------
// Host launch
// ---------------------------------------------------------------------------
extern "C" void kernel_launch(void* const* d_in, const int* in_sizes, int n_in,
                              void* d_out, int out_size, void* d_ws, size_t ws_size,
                              hipStream_t stream) {
  // setup_inputs() order:
  const float* x     = (const float*)d_in[0];
  const int*   idx_i = (const int*)  d_in[1];
  const int*   idx_j = (const int*)  d_in[2];
  const float* Wq = (const float*)d_in[3];  const float* bq = (const float*)d_in[4];
  const float* Wk = (const float*)d_in[5];  const float* bk = (const float*)d_in[6];
  const float* Wv = (const float*)d_in[7];  const float* bv = (const float*)d_in[8];
  const float* Wo = (const float*)d_in[9];  const float* bo = (const float*)d_in[10];
  const float* g1 = (const float*)d_in[11]; const float* be1 = (const float*)d_in[12];
  const float* W1 = (const float*)d_in[13]; const float* bb1 = (const float*)d_in[14];
  const float* W2 = (const float*)d_in[15]; const float* bb2 = (const float*)d_in[16];
  const float* g2 = (const float*)d_in[17]; const float* be2 = (const float*)d_in[18];
  float* outp = (float*)d_out;

  // workspace carving (floats)
  float* ws = (float*)d_ws;
  const size_t ND = (size_t)NN * DDIM;   // 6.4M
  const size_t EH = (size_t)EE * HH;     // 6.4M
  const size_t NH = (size_t)NN * HH;     // 0.4M
  float* q     = ws;            ws += ND;
  float* k     = ws;            ws += ND;
  float* v     = ws;            ws += ND;
  float* e     = ws;            ws += EH;
  float* m     = ws;            ws += NH;
  float* denom = ws;            ws += NH;
  float* agg   = ws;            ws += ND;
  float* h     = ws;            ws += ND;   // total ~157 MB

  const int ethreads = EE * HH;

  k_init<<<(int)((ND + 255) / 256), 256, 0, stream>>>(m, denom, agg);
  k_qkv<<<TILES, 256, 0, stream>>>(x, Wq, bq, Wk, bk, Wv, bv, q, k, v);
  k_scores<<<(ethreads + 255) / 256, 256, 0, stream>>>(q, k, idx_i, idx_j, e, m);
  k_accum<<<(ethreads + 255) / 256, 256, 0, stream>>>(e, m, v, idx_i, idx_j, denom, agg);
  k_attn_out<<<TILES, 256, 0, stream>>>(x, agg, denom, Wo, bo, g1, be1, h);
  k_ffn<<<TILES, 256, 0, stream>>>(h, W1, bb1, W2, bb2, g2, be2, outp);
}